// LearnableFeatureFilter_70703751627163
// MI455X (gfx1250) — compile-verified
//
#include <hip/hip_runtime.h>
#include <math.h>

// ---------------------------------------------------------------------------
// LearnableFeatureFilter for MI455X (gfx1250, wave32).
// x: (B=16, C=256, H=64, W=64) fp32.  Heavy einsum done with
// V_WMMA_F32_16X16X4_F32 (fp32 matrix core), exact 0.85-quantile via 4-round
// radix select on |x| bit patterns, everything else fused/memory-lean.
// ---------------------------------------------------------------------------

#define B 16
#define C 256
#define R 16
#define H 64
#define W 64
#define HW (H*W)          // 4096
#define NE ((size_t)C*HW) // 1048576 elements per batch

typedef __attribute__((ext_vector_type(2))) float v2f;
typedef __attribute__((ext_vector_type(8))) float v8f;

__device__ __forceinline__ float sigmoidf_(float v) {
    return 1.0f / (1.0f + __expf(-v));
}

// ---------------- Stage 1: global average pool --------------------------
// one block per (b,c); 256 threads reduce 4096 elements
__global__ __launch_bounds__(256) void k_pooled(const float* __restrict__ x,
                                                float* __restrict__ pooled) {
    __shared__ float red[256];
    int t = threadIdx.x;
    const float* p = x + (size_t)blockIdx.x * HW;
    float s = 0.f;
    for (int i = t; i < HW; i += 256) s += p[i];
    red[t] = s;
    __syncthreads();
    for (int w = 128; w > 0; w >>= 1) {
        if (t < w) red[t] += red[t + w];
        __syncthreads();
    }
    if (t == 0) pooled[blockIdx.x] = red[0] * (1.0f / (float)HW);
}

// ---------------- Stage 2: channel-attention MLP (tiny) ------------------
__global__ __launch_bounds__(256) void k_ca(const float* __restrict__ pooled,
                                            const float* __restrict__ w1,
                                            const float* __restrict__ b1,
                                            const float* __restrict__ w2,
                                            const float* __restrict__ b2,
                                            float* __restrict__ cw) {
    __shared__ float sh[B * R];
    int t = threadIdx.x;
    {   // h[b][r] = relu(pooled[b,:] . w1[r,:] + b1[r])
        int bb = t >> 4, r = t & 15;
        float acc = b1[r];
        const float* pb = pooled + bb * C;
        const float* wr = w1 + r * C;
        for (int c = 0; c < C; ++c) acc += pb[c] * wr[c];
        sh[t] = acc > 0.f ? acc : 0.f;
    }
    __syncthreads();
    {   // cw[b][c] = sigmoid(h[b,:] . w2[c,:] + b2[c])
        int c = t;
        float bias = b2[c];
        const float* wc = w2 + c * R;
        for (int bb = 0; bb < B; ++bb) {
            float acc = bias;
            const float* hb = sh + bb * R;
            #pragma unroll
            for (int r = 0; r < R; ++r) acc += hb[r] * wc[r];
            cw[bb * C + c] = sigmoidf_(acc);
        }
    }
}

// ---------------- Stage 3: exact 0.85-quantile of |x| (radix select) ------
// state: prefix[b][s], rank[b][s] for s in {0,1} (ranks k and k+1)
__global__ __launch_bounds__(256) void k_qinit(unsigned* __restrict__ prefix,
                                               unsigned* __restrict__ rank,
                                               unsigned* __restrict__ hist,
                                               unsigned k0) {
    int t = threadIdx.x;
    for (int i = t; i < B * 2 * 256; i += 256) hist[i] = 0u;
    if (t < B * 2) { prefix[t] = 0u; rank[t] = k0 + (unsigned)(t & 1); }
}

// 32 chunk-blocks per batch; LDS histograms for both rank-states
__global__ __launch_bounds__(256) void k_qhist(const float* __restrict__ x,
                                               const unsigned* __restrict__ prefix,
                                               unsigned* __restrict__ hist,
                                               int shift) {
    __shared__ unsigned lh[512];
    int t = threadIdx.x;
    int b = blockIdx.x >> 5;
    int chunk = blockIdx.x & 31;
    for (int i = t; i < 512; i += 256) lh[i] = 0u;
    __syncthreads();
    unsigned p0 = prefix[b * 2 + 0];
    unsigned p1 = prefix[b * 2 + 1];
    unsigned maskhi = (shift == 24) ? 0u : (0xFFFFFFFFu << (shift + 8));
    const float* xb = x + (size_t)b * NE;
    size_t start = (size_t)chunk * (NE / 32);
    size_t end = start + (NE / 32);
    for (size_t i = start + t; i < end; i += 256) {
        unsigned u = __float_as_uint(xb[i]) & 0x7FFFFFFFu; // |x| bits (monotone)
        unsigned bin = (u >> shift) & 0xFFu;
        if ((u & maskhi) == p0) atomicAdd(&lh[bin], 1u);
        if ((u & maskhi) == p1) atomicAdd(&lh[256 + bin], 1u);
    }
    __syncthreads();
    for (int i = t; i < 512; i += 256)
        if (lh[i]) atomicAdd(&hist[(b * 2 + (i >> 8)) * 256 + (i & 255)], lh[i]);
}

__global__ __launch_bounds__(32) void k_qpick(unsigned* __restrict__ prefix,
                                              unsigned* __restrict__ rank,
                                              unsigned* __restrict__ hist,
                                              int shift) {
    int t = threadIdx.x;           // t = b*2 + s
    if (t >= B * 2) return;
    unsigned* h = hist + t * 256;
    unsigned r = rank[t];
    unsigned cum = 0, sel = 255u;
    for (int bin = 0; bin < 256; ++bin) {
        unsigned c = h[bin];
        if (cum + c > r) { sel = (unsigned)bin; r -= cum; break; }
        cum += c;
    }
    prefix[t] |= sel << shift;
    rank[t] = r;
    for (int bin = 0; bin < 256; ++bin) h[bin] = 0u; // ready for next round
}

__global__ __launch_bounds__(32) void k_qfin(const unsigned* __restrict__ prefix,
                                             float* __restrict__ gthr, float frac) {
    int t = threadIdx.x;
    if (t < B) {
        float lo = __uint_as_float(prefix[t * 2 + 0]);
        float hi = __uint_as_float(prefix[t * 2 + 1]);
        gthr[t] = (1.0f - frac) * lo + frac * hi;
    }
}

// ---------------- Stage 4: 3x3 conv (C->1) + sigmoid -> spatial weights ---
// grid: (B * 16) blocks; block handles 4 rows x 64 cols; weights in LDS
__global__ __launch_bounds__(256) void k_conv(const float* __restrict__ x,
                                              const float* __restrict__ sa_w,
                                              const float* __restrict__ sa_b,
                                              float* __restrict__ sw) {
    __shared__ float wt[C * 9];
    int t = threadIdx.x;
    for (int i = t; i < C * 9; i += 256) wt[i] = sa_w[i];
    __syncthreads();
    int b = blockIdx.x >> 4;
    int hg = blockIdx.x & 15;
    int h = hg * 4 + (t >> 6);
    int w = t & 63;
    const float* xb = x + (size_t)b * NE;
    float acc = 0.f;
    for (int c = 0; c < C; ++c) {
        const float* xc = xb + (size_t)c * HW;
        const float* wc = wt + c * 9;
        #pragma unroll
        for (int dy = -1; dy <= 1; ++dy) {
            int hh = h + dy;
            if (hh < 0 || hh >= H) continue;
            const float* row = xc + hh * W;
            #pragma unroll
            for (int dx = -1; dx <= 1; ++dx) {
                int ww = w + dx;
                if (ww < 0 || ww >= W) continue;
                acc += row[ww] * wc[(dy + 1) * 3 + (dx + 1)];
            }
        }
    }
    sw[(size_t)b * HW + h * W + w] = sigmoidf_(acc + sa_b[0]);
}

// ---------------- Stage 5: fused WMMA einsum + threshold + output ---------
// Per wave: one 16(R) x 16(pos) tile via V_WMMA_F32_16X16X4_F32, K-loop over
// C=256.  Then 1x16 projection + sigmoid -> relative threshold, then final
// elementwise over all 256 channels at those 16 positions.
__global__ __launch_bounds__(256) void k_fused(const float* __restrict__ x,
                                               const float* __restrict__ tp_w1,
                                               const float* __restrict__ tp_b1,
                                               const float* __restrict__ tp_w2,
                                               const float* __restrict__ tp_b2,
                                               const float* __restrict__ cw,
                                               const float* __restrict__ sw,
                                               const float* __restrict__ gthr,
                                               float* __restrict__ out) {
    __shared__ float sW1[R * C];   // 16 KB
    __shared__ float sCW[C];
    __shared__ float sB1[R], sW2[R];
    int t = threadIdx.x;
    int b  = blockIdx.x >> 5;      // 32 blocks per batch
    int tb = blockIdx.x & 31;
    for (int i = t; i < R * C; i += 256) sW1[i] = tp_w1[i];
    if (t < C)  sCW[t] = cw[b * C + t];
    if (t < R) { sB1[t] = tp_b1[t]; sW2[t] = tp_w2[t]; }
    __syncthreads();

    int lane = t & 31, wv = t >> 5;
    int tile = tb * 8 + wv;        // 0..255
    int pos0 = tile << 4;
    const float* xb = x + (size_t)b * NE;

    // fragment lane mapping (ISA 7.12.2, wave32, f32 16x16x4):
    //   A: M = lane&15, K = {0,1} (lanes 0-15) / {2,3} (lanes 16-31)
    //   B: N = lane&15, same K split
    int n  = lane & 15;
    int kk = (lane >> 4) << 1;     // 0 or 2
    v8f acc = {};
    for (int k0 = 0; k0 < C; k0 += 4) {
        v2f a, bf;
        a.x  = sW1[n * C + k0 + kk];
        a.y  = sW1[n * C + k0 + kk + 1];
        bf.x = xb[(size_t)(k0 + kk)     * HW + pos0 + n];
        bf.y = xb[(size_t)(k0 + kk + 1) * HW + pos0 + n];
        acc = __builtin_amdgcn_wmma_f32_16x16x4_f32(
                  false, a, false, bf, (short)0, acc, false, false);
    }

    // D layout: element r <-> M = r + 8*(lane>=16), N = lane&15
    int mbase = (lane >> 4) << 3;
    float partial = 0.f;
    #pragma unroll
    for (int r = 0; r < 8; ++r) {
        int m = mbase + r;
        float thv = acc[r] + sB1[m];
        thv = thv > 0.f ? thv : 0.f;   // relu
        partial += sW2[m] * thv;
    }
    partial += __shfl_xor(partial, 16, 32);    // combine M halves
    float rel = sigmoidf_(partial + tp_b2[0]);
    float thr = gthr[b] * rel;                 // absolute threshold @ this pos
    float swv = sw[(size_t)b * HW + pos0 + n]; // spatial weight @ this pos

    float* ob = out + (size_t)b * NE;
    for (int c0 = 0; c0 < C; c0 += 2) {
        int c = c0 + (lane >> 4);
        size_t idx = (size_t)c * HW + pos0 + n;
        float xv = xb[idx];
        float mk = sigmoidf_((fabsf(xv) - thr) * 10.0f);
        ob[idx] = xv * mk * sCW[c] * swv;
    }
}

// ---------------------------------------------------------------------------
extern "C" void kernel_launch(void* const* d_in, const int* in_sizes, int n_in,
                              void* d_out, int out_size, void* d_ws, size_t ws_size,
                              hipStream_t stream) {
    (void)in_sizes; (void)n_in; (void)out_size; (void)ws_size;
    const float* x     = (const float*)d_in[0];
    const float* ca_w1 = (const float*)d_in[1];
    const float* ca_b1 = (const float*)d_in[2];
    const float* ca_w2 = (const float*)d_in[3];
    const float* ca_b2 = (const float*)d_in[4];
    const float* sa_w  = (const float*)d_in[5];
    const float* sa_b  = (const float*)d_in[6];
    const float* tp_w1 = (const float*)d_in[7];
    const float* tp_b1 = (const float*)d_in[8];
    const float* tp_w2 = (const float*)d_in[9];
    const float* tp_b2 = (const float*)d_in[10];
    float* out = (float*)d_out;

    // workspace layout (float-element offsets)
    float* ws      = (float*)d_ws;
    float* pooled  = ws;                 // 4096
    float* cw      = ws + 4096;          // 4096
    float* sw      = ws + 8192;          // 65536
    float* gthr    = ws + 73728;         // 16
    unsigned* prefix = (unsigned*)(ws + 73744); // 32
    unsigned* rank   = prefix + 32;             // 32
    unsigned* hist   = rank + 32;               // 16*2*256 = 8192

    // quantile rank: q = 0.85 * (NE-1), linear interpolation
    double q = 0.85 * (double)(NE - 1);
    unsigned k0 = (unsigned)q;
    float frac = (float)(q - (double)k0);

    k_pooled<<<B * C, 256, 0, stream>>>(x, pooled);
    k_ca<<<1, 256, 0, stream>>>(pooled, ca_w1, ca_b1, ca_w2, ca_b2, cw);

    k_qinit<<<1, 256, 0, stream>>>(prefix, rank, hist, k0);
    for (int shift = 24; shift >= 0; shift -= 8) {
        k_qhist<<<B * 32, 256, 0, stream>>>(x, prefix, hist, shift);
        k_qpick<<<1, 32, 0, stream>>>(prefix, rank, hist, shift);
    }
    k_qfin<<<1, 32, 0, stream>>>(prefix, gthr, frac);

    k_conv<<<B * 16, 256, 0, stream>>>(x, sa_w, sa_b, sw);

    k_fused<<<B * 32, 256, 0, stream>>>(x, tp_w1, tp_b1, tp_w2, tp_b2,
                                        cw, sw, gthr, out);
}